// IMPACTModel_low_mem_21234318311842
// MI455X (gfx1250) — compile-verified
//
#include <hip/hip_runtime.h>

typedef __attribute__((ext_vector_type(2))) float v2f;
typedef __attribute__((ext_vector_type(8))) float v8f;

#define M_DIM 14      // NB_MOD_MAX + 2
#define C_DIM 128
#define D_IN  3

// One wave32 per batch element.
//   A (16x4 f32, negated a-rows, K=3 and M=14 zero-padded)
//   B tile t (4x16 f32) = W_t[:, t*16 .. t*16+15], K=3 zero-padded
//   C init  = u[t*16 + n] broadcast over rows  =>  D = u - a*W_t = diff
__global__ __launch_bounds__(256) void impact_wmma_kernel(
    const int*   __restrict__ user_ids,
    const int*   __restrict__ item_ids,
    const int*   __restrict__ concept_ids,
    const float* __restrict__ users_emb,   // [USER_N][128]
    const float* __restrict__ item_resp,   // [ITEM_N*14][3]
    const float* __restrict__ W,           // [128][3][128]
    const float* __restrict__ mask,        // [ITEM_N][14]
    const int*   __restrict__ nb_mod,      // [ITEM_N]
    float*       __restrict__ out,
    int B)
{
    const int lane = threadIdx.x & 31;
    const int wave = (int)((blockIdx.x * blockDim.x + threadIdx.x) >> 5);
    if (wave >= B) return;               // wave-uniform; EXEC stays all-ones for WMMA

    const int uid = user_ids[wave];
    const int iid = item_ids[wave];
    const int cid = concept_ids[wave];

    const float* u  = users_emb + (size_t)uid * C_DIM;
    const float* Wt = W         + (size_t)cid * (D_IN * C_DIM);
    const float* ir = item_resp + (size_t)iid * (M_DIM * D_IN);

    const bool lo  = lane < 16;
    const int  r16 = lane & 15;
    const bool row_ok = r16 < M_DIM;
    const int  rclmp  = row_ok ? r16 : (M_DIM - 1);   // clamp keeps addresses in-bounds

    // ---- A matrix, negated; branchless (clamped loads + value select).
    // lane L (low half): row L holds K=0/1 ; high half: row L-16 holds K=2 (K=3 pad = 0)
    {
        const float va = ir[rclmp * 3 + (lo ? 0 : 2)];   // address select, single load
        const float vb = ir[rclmp * 3 + 1];              // unconditional load
        v2f Atmp;
        Atmp.x = row_ok ? -va : 0.0f;
        Atmp.y = (row_ok && lo) ? -vb : 0.0f;
        // keep name A for the MMA operand
        v2f A = Atmp;

        // ---- u: 4 fully-coalesced dword loads per lane, redistributed by shuffle per tile
        float ureg[4];
        #pragma unroll
        for (int k = 0; k < 4; ++k) ureg[k] = u[k * 32 + lane];

        v8f s;
        #pragma unroll
        for (int r = 0; r < 8; ++r) s[r] = 0.0f;

        const int bxbase = lo ? 0 : 2 * C_DIM;   // one cndmask, hoisted out of the loop

        #pragma unroll
        for (int t = 0; t < 8; ++t) {
            const int n = t * 16 + r16;
            // B tile: VGPR0 -> K = (lo?0:2), VGPR1 -> K = (lo?1:3=pad0), N = n
            v2f Bt;
            Bt.x = Wt[bxbase + n];               // unconditional, address pre-selected
            const float w1 = Wt[C_DIM + n];      // unconditional load
            Bt.y = lo ? w1 : 0.0f;               // value select, no EXEC branch
            // C init: u[n] broadcast to all 16 rows (same value in all 8 acc VGPRs)
            const float uval = __shfl(ureg[t >> 1], (t & 1) * 16 + r16, 32);
            v8f c;
            #pragma unroll
            for (int r = 0; r < 8; ++r) c[r] = uval;

            v8f d = __builtin_amdgcn_wmma_f32_16x16x4_f32(
                /*neg_a=*/false, A, /*neg_b=*/false, Bt,
                /*c_mod=*/(short)0, c, /*reuse_a=*/false, /*reuse_b=*/false);

            #pragma unroll
            for (int r = 0; r < 8; ++r) s[r] += d[r] * d[r];   // diff^2 accumulate
        }

        // ---- reduce over the 16 N-lanes of each half (xor masks 1,2,4,8 stay within a half)
        #pragma unroll
        for (int off = 1; off < 16; off <<= 1) {
            #pragma unroll
            for (int r = 0; r < 8; ++r) s[r] += __shfl_xor((float)s[r], off, 32);
        }
        // now: s[r] == p_uim[row] with row = r + (lo ? 0 : 8), uniform within each half

        const float INF = __builtin_inff();
        const int   rowbase = lo ? 0 : 8;
        float bestv = INF;
        int   besti = 0;
        #pragma unroll
        for (int r = 0; r < 8; ++r) {
            const int   row  = r + rowbase;
            const int   rowc = (row < M_DIM) ? row : (M_DIM - 1);     // clamp address
            const float mraw = mask[(size_t)iid * M_DIM + rowc];      // unconditional load
            const float mv   = (row < M_DIM) ? mraw : INF;            // value select
            const float v    = s[r] + mv;
            if (v < bestv) { bestv = v; besti = row; }   // ascending scan => first min
        }
        // combine the two halves with first-index tie-break
        const float ov = __shfl_xor(bestv, 16, 32);
        const int   oi = __shfl_xor(besti, 16, 32);
        if (ov < bestv || (ov == bestv && oi < besti)) { bestv = ov; besti = oi; }

        if (lane == 0) {
            const float nb = (float)nb_mod[iid];
            out[wave] = ((float)besti - 1.0f) / (nb - 1.0f) + 1.0f;
        }
    }
}

extern "C" void kernel_launch(void* const* d_in, const int* in_sizes, int n_in,
                              void* d_out, int out_size, void* d_ws, size_t ws_size,
                              hipStream_t stream) {
    (void)n_in; (void)out_size; (void)d_ws; (void)ws_size;
    const int*   user_ids    = (const int*)  d_in[0];
    const int*   item_ids    = (const int*)  d_in[1];
    const int*   concept_ids = (const int*)  d_in[2];
    const float* users_emb   = (const float*)d_in[3];
    const float* item_resp   = (const float*)d_in[4];
    const float* W           = (const float*)d_in[5];
    const float* mask        = (const float*)d_in[6];
    const int*   nb_mod      = (const int*)  d_in[7];
    float*       out         = (float*)d_out;

    const int B = in_sizes[0];                    // 131072
    const int threads = 256;                      // 8 waves per block
    const long long total_threads = (long long)B * 32;
    const int blocks = (int)((total_threads + threads - 1) / threads);

    impact_wmma_kernel<<<blocks, threads, 0, stream>>>(
        user_ids, item_ids, concept_ids, users_emb, item_resp, W, mask, nb_mod, out, B);
}